// QuantumTransformerBlock_65481071399148
// MI455X (gfx1250) — compile-verified
//
#include <hip/hip_runtime.h>
#include <math.h>

#define EPSV 1e-5f
constexpr int BQ = 1024;   // x batch dim (== rows of attn/ffn, broadcast over sequence)
constexpr int SQ = 1024;   // sequence length
constexpr int E  = 12;     // embedding dim
constexpr int NQ = 12;     // qubits

typedef float v2f __attribute__((ext_vector_type(2)));
typedef float v4f __attribute__((ext_vector_type(4)));
typedef float v8f __attribute__((ext_vector_type(8)));

// ---------------- Kernel 1: attn[s,q] (closed form of circuit 1) ----------------
// z_i = cos(rx[i]) * cos(x[s,i,0]);  attn[s,0]=prod_{i>=1} z_i; attn[s,k]=prod_{i<=k} z_i
__global__ void k_attn(const float* __restrict__ x, const float* __restrict__ rx,
                       float* __restrict__ attn) {
  int s = blockIdx.x * blockDim.x + threadIdx.x;
  if (s >= BQ) return;
  float z[NQ];
#pragma unroll
  for (int i = 0; i < NQ; ++i)
    z[i] = cosf(rx[i]) * cosf(x[((size_t)s * SQ + i) * E]);
  float p1 = 1.f;
#pragma unroll
  for (int i = 1; i < NQ; ++i) p1 *= z[i];
  attn[s * NQ + 0] = p1;
  float p = z[0];
#pragma unroll
  for (int k = 1; k < NQ; ++k) { p *= z[k]; attn[s * NQ + k] = p; }
}

// ---------------- Kernel 2a: rz[s,q] = relu(cos(ry[q] + x1[s,q,0])) ----------------
__global__ void k_rz(const float* __restrict__ x, const float* __restrict__ attn,
                     const float* __restrict__ ry, const float* __restrict__ g1,
                     const float* __restrict__ b1, float* __restrict__ rz) {
  int idx = blockIdx.x * blockDim.x + threadIdx.x;
  if (idx >= BQ * NQ) return;
  int s = idx / NQ, q = idx - s * NQ;
  const float* row = x + ((size_t)s * SQ + q) * E;
  float y[E];
  float sum = 0.f;
#pragma unroll
  for (int k = 0; k < E; ++k) { y[k] = row[k] + attn[q * E + k]; sum += y[k]; }
  float mu = sum * (1.f / E);
  float var = 0.f;
#pragma unroll
  for (int k = 0; k < E; ++k) { float d = y[k] - mu; var += d * d; }
  var *= (1.f / E);
  float x10 = (y[0] - mu) * rsqrtf(var + EPSV) * g1[0] + b1[0];
  rz[idx] = fmaxf(cosf(ry[q] + x10), 0.f);
}

// ---------------- Kernel 2b: ffn = rz @ w_ffn^T + b  via V_WMMA_F32_16X16X4_F32 ----
// One wave computes a 16(batch) x 16(out, 12 valid) tile; K=12 in 3 chunks of 4.
// A (16x4 f32): lanes 0-15 hold K={k0,k0+1}, lanes 16-31 hold K={k0+2,k0+3} (ISA 7.12.2)
// B (4x16 f32): mirrored row striping.  C/D: vgpr v -> row v (lanes<16) / v+8 (lanes>=16)
// Padding lanes (n=12..15) use clamped in-bounds loads zeroed by a mask multiply,
// so no exec-mask branching occurs and EXEC stays all-ones through the WMMAs.
__global__ void k_ffn_wmma(const float* __restrict__ rz, const float* __restrict__ w,
                           const float* __restrict__ bf, float* __restrict__ ffn) {
  int gtid = blockIdx.x * blockDim.x + threadIdx.x;
  int wave = gtid >> 5;
  int lane = threadIdx.x & 31;
  int sbase = wave * 16;
  int half = lane >> 4;   // 0: lanes 0-15, 1: lanes 16-31
  int m    = lane & 15;   // A row (M) for loads, also D column (N) index
  int nc   = (m < E) ? m : (E - 1);       // clamped B column (always in-bounds)
  float msk = (m < E) ? 1.f : 0.f;        // zero-out padding columns
  v8f c = {0.f, 0.f, 0.f, 0.f, 0.f, 0.f, 0.f, 0.f};
#pragma unroll
  for (int chunk = 0; chunk < 3; ++chunk) {
    int k0 = chunk * 4;
    int ka = k0 + half * 2;
    v2f a, b;
    a.x = rz[(sbase + m) * NQ + ka];
    a.y = rz[(sbase + m) * NQ + ka + 1];
    b.x = w[nc * NQ + ka]     * msk;      // B[k][n] = w_ffn[n][k]
    b.y = w[nc * NQ + ka + 1] * msk;
    c = __builtin_amdgcn_wmma_f32_16x16x4_f32(false, a, false, b, (short)0, c,
                                              false, false);
  }
  if (m < E) {
#pragma unroll
    for (int v = 0; v < 8; ++v) {
      int mr = v + half * 8;
      ffn[(sbase + mr) * E + m] = c[v] + bf[m];
    }
  }
}

// ---------------- Kernel 3: fused double-LayerNorm stream ----------------
// out[i,j,:] = LN2( LN1(x[i,j,:] + attn[j,:]) + ffn[j,:] )
__global__ void __launch_bounds__(256)
k_main(const float* __restrict__ x, const float* __restrict__ attn,
       const float* __restrict__ ffn,
       const float* __restrict__ g1, const float* __restrict__ b1,
       const float* __restrict__ g2, const float* __restrict__ b2,
       float* __restrict__ out) {
  int r = blockIdx.x * blockDim.x + threadIdx.x;
  if (r >= BQ * SQ) return;
  int j = r & (SQ - 1);  // sequence index

  const v4f* xr = (const v4f*)(x + (size_t)r * E);
  v4f v0 = __builtin_nontemporal_load(xr + 0);
  v4f v1 = __builtin_nontemporal_load(xr + 1);
  v4f v2 = __builtin_nontemporal_load(xr + 2);

  float y[E] = {v0.x, v0.y, v0.z, v0.w, v1.x, v1.y, v1.z, v1.w,
                v2.x, v2.y, v2.z, v2.w};
  const float* ar = attn + j * E;
  const float* fr = ffn  + j * E;

  float sum = 0.f;
#pragma unroll
  for (int k = 0; k < E; ++k) { y[k] += ar[k]; sum += y[k]; }
  float mu = sum * (1.f / E);
  float var = 0.f;
#pragma unroll
  for (int k = 0; k < E; ++k) { float d = y[k] - mu; var += d * d; }
  float rs = rsqrtf(var * (1.f / E) + EPSV);

  float t[E];
  float sum2 = 0.f;
#pragma unroll
  for (int k = 0; k < E; ++k) {
    t[k] = (y[k] - mu) * rs * g1[k] + b1[k] + fr[k];
    sum2 += t[k];
  }
  float mu2 = sum2 * (1.f / E);
  float var2 = 0.f;
#pragma unroll
  for (int k = 0; k < E; ++k) { float d = t[k] - mu2; var2 += d * d; }
  float rs2 = rsqrtf(var2 * (1.f / E) + EPSV);

  v4f o0, o1, o2;
  float o[E];
#pragma unroll
  for (int k = 0; k < E; ++k) o[k] = (t[k] - mu2) * rs2 * g2[k] + b2[k];
  o0.x = o[0];  o0.y = o[1];  o0.z = o[2];  o0.w = o[3];
  o1.x = o[4];  o1.y = o[5];  o1.z = o[6];  o1.w = o[7];
  o2.x = o[8];  o2.y = o[9];  o2.z = o[10]; o2.w = o[11];

  v4f* orow = (v4f*)(out + (size_t)r * E);
  __builtin_nontemporal_store(o0, orow + 0);
  __builtin_nontemporal_store(o1, orow + 1);
  __builtin_nontemporal_store(o2, orow + 2);
}

extern "C" void kernel_launch(void* const* d_in, const int* in_sizes, int n_in,
                              void* d_out, int out_size, void* d_ws, size_t ws_size,
                              hipStream_t stream) {
  const float* x    = (const float*)d_in[0];
  const float* rx   = (const float*)d_in[1];
  const float* ry   = (const float*)d_in[2];
  const float* wf   = (const float*)d_in[3];
  const float* bfn  = (const float*)d_in[4];
  const float* g1   = (const float*)d_in[5];
  const float* b1   = (const float*)d_in[6];
  const float* g2   = (const float*)d_in[7];
  const float* b2   = (const float*)d_in[8];
  float* out = (float*)d_out;

  float* attn = (float*)d_ws;            // 1024*12 floats
  float* rz   = attn + BQ * NQ;          // 1024*12 floats
  float* ffn  = rz   + BQ * NQ;          // 1024*12 floats

  // 1) attn (closed-form circuit 1)
  k_attn<<<(BQ + 255) / 256, 256, 0, stream>>>(x, rx, attn);
  // 2a) rz = relu(cos(ry + x1[:, :12, 0]))
  k_rz<<<(BQ * NQ + 255) / 256, 256, 0, stream>>>(x, attn, ry, g1, b1, rz);
  // 2b) ffn = rz @ w^T + b  (64 waves, 16 rows each, f32 WMMA)
  k_ffn_wmma<<<(BQ / 16) * 32 / 256, 256, 0, stream>>>(rz, wf, bfn, ffn);
  // 3) fused double-LN stream over 1M rows
  k_main<<<(BQ * SQ + 255) / 256, 256, 0, stream>>>(x, attn, ffn, g1, b1, g2, b2, out);
}